// VNEdgeConv_59021440582192
// MI455X (gfx1250) — compile-verified
//
#include <hip/hip_runtime.h>
#include <hip/hip_bf16.h>

typedef __attribute__((ext_vector_type(2))) float v2f;
typedef __attribute__((ext_vector_type(8))) float v8f;

#define B_ 4
#define N_ 2048
#define K_ 20

// ---------------- Kernel 1: pos = mean_D(x)  (B,N,3) ----------------
__global__ void pos_kernel(const float* __restrict__ x, float* __restrict__ pos) {
    int i = blockIdx.x * blockDim.x + threadIdx.x;   // (b*N+n)*3+v
    if (i >= B_ * N_ * 3) return;
    const float4* p = (const float4*)(x + (size_t)i * 64);
    float s = 0.f;
#pragma unroll
    for (int j = 0; j < 16; ++j) {
        float4 t = p[j];
        s += t.x + t.y + t.z + t.w;
    }
    pos[i] = s * (1.0f / 64.0f);
}

// ---------------- Kernel 2: exact top-20 nearest neighbors ----------------
// 64 blocks x 128 threads; each block: 128 queries of one batch.
__global__ void __launch_bounds__(128) knn_kernel(const float* __restrict__ pos,
                                                  int* __restrict__ knn) {
    __shared__ float sp[N_ * 3];
    __shared__ float bd[K_][128];
    __shared__ int   bi[K_][128];
    int b = blockIdx.x >> 4;
    int qbase = (blockIdx.x & 15) << 7;
    int t = threadIdx.x;
    for (int e = t; e < N_ * 3; e += 128) sp[e] = pos[b * N_ * 3 + e];
    __syncthreads();
    int q = qbase + t;
    float qx = sp[q * 3], qy = sp[q * 3 + 1], qz = sp[q * 3 + 2];
#pragma unroll
    for (int kk = 0; kk < K_; ++kk) bd[kk][t] = 3.0e38f;
    float worst = 3.0e38f;
    int wpos = 0;
    for (int j = 0; j < N_; ++j) {
        float dx = sp[j * 3] - qx;
        float dy = sp[j * 3 + 1] - qy;
        float dz = sp[j * 3 + 2] - qz;
        float d = dx * dx + dy * dy + dz * dz;
        if (j != q && d < worst) {
            bd[wpos][t] = d;
            bi[wpos][t] = j;
            worst = -1.0f;
#pragma unroll
            for (int kk = 0; kk < K_; ++kk) {
                float v = bd[kk][t];
                if (v > worst) { worst = v; wpos = kk; }
            }
        }
    }
    for (int kk = 0; kk < K_; ++kk) knn[((size_t)b * N_ + q) * K_ + kk] = bi[kk][t];
}

// ---------------- Kernel 3: fused edge->W1->VNrelu->W2->argmax ----------------
// One block of 128 threads (4 wave32) per (b,n). Rows r = 3k+v, 60 rows + 4 zero pad.
__global__ void __launch_bounds__(128) fused_kernel(const float* __restrict__ x,
                                                    const float* __restrict__ W1,
                                                    const float* __restrict__ W2,
                                                    const int* __restrict__ knn,
                                                    float* __restrict__ out) {
    __shared__ float sE[64 * 128];   // edge (phase 1); first 64*64 reused as h2 (phase 3)
    __shared__ float sH[64 * 64];    // h1 / vn-relu'd h1
    __shared__ float sXS[3 * 64];    // x self
    __shared__ int   sIdx[K_];

    int t = threadIdx.x;
    int bn = blockIdx.x;                               // b*N + n
    const float* xb = x + (size_t)(bn / N_) * N_ * 192; // batch base
    const float* xs = x + (size_t)bn * 192;             // this point (3x64)

    for (int e = t; e < 192; e += 128) sXS[e] = xs[e];
    if (t < K_) sIdx[t] = knn[(size_t)bn * K_ + t];
    __syncthreads();

    // ---- build edge[64][128]: cols 0..63 = x_self, 64..127 = nbr - x_self ----
    for (int e = t; e < 64 * 128; e += 128) {
        int r = e >> 7;
        int d = e & 127;
        float val = 0.f;
        if (r < 60) {
            int k = r / 3;
            int v = r - 3 * k;
            if (d < 64) {
                val = sXS[v * 64 + d];
            } else {
                int dd = d - 64;
                val = xb[(size_t)sIdx[k] * 192 + v * 64 + dd] - sXS[v * 64 + dd];
            }
        }
        sE[e] = val;
    }
    __syncthreads();

    int wave = t >> 5;
    int lane = t & 31;
    int lh = lane >> 4;     // lane half (0/1)
    int lm = lane & 15;
    int m0 = wave * 16;

    // ---- matmul1: h1[64][64] = edge[64][128] @ W1[128][64], fp32 WMMA ----
    for (int nt = 0; nt < 4; ++nt) {
        int n0 = nt * 16;
        v8f acc = {};
#pragma unroll 4
        for (int ks = 0; ks < 32; ++ks) {
            int ka = ks * 4 + lh * 2;                      // K pair base for this lane half
            v2f a, bf;
            a.x = sE[(m0 + lm) * 128 + ka];
            a.y = sE[(m0 + lm) * 128 + ka + 1];
            bf.x = W1[ka * 64 + n0 + lm];
            bf.y = W1[(ka + 1) * 64 + n0 + lm];
            acc = __builtin_amdgcn_wmma_f32_16x16x4_f32(false, a, false, bf,
                                                        (short)0, acc, false, false);
        }
#pragma unroll
        for (int r = 0; r < 8; ++r)
            sH[(m0 + r + lh * 8) * 64 + n0 + lm] = acc[r];
    }
    __syncthreads();

    // ---- VN relu: scale each channel's 3-vector by n/(n+1e-8) ----
    for (int p = t; p < K_ * 64; p += 128) {
        int k = p >> 6;
        int c = p & 63;
        float h0 = sH[(3 * k) * 64 + c];
        float h1 = sH[(3 * k + 1) * 64 + c];
        float h2 = sH[(3 * k + 2) * 64 + c];
        float nrm = sqrtf(h0 * h0 + h1 * h1 + h2 * h2);
        float f = nrm / (nrm + 1e-8f);
        sH[(3 * k) * 64 + c] = h0 * f;
        sH[(3 * k + 1) * 64 + c] = h1 * f;
        sH[(3 * k + 2) * 64 + c] = h2 * f;
    }
    __syncthreads();

    // ---- matmul2: h2[64][64] = h1'[64][64] @ W2[64][64] (into sE region) ----
    for (int nt = 0; nt < 4; ++nt) {
        int n0 = nt * 16;
        v8f acc = {};
#pragma unroll 4
        for (int ks = 0; ks < 16; ++ks) {
            int ka = ks * 4 + lh * 2;
            v2f a, bf;
            a.x = sH[(m0 + lm) * 64 + ka];
            a.y = sH[(m0 + lm) * 64 + ka + 1];
            bf.x = W2[ka * 64 + n0 + lm];
            bf.y = W2[(ka + 1) * 64 + n0 + lm];
            acc = __builtin_amdgcn_wmma_f32_16x16x4_f32(false, a, false, bf,
                                                        (short)0, acc, false, false);
        }
#pragma unroll
        for (int r = 0; r < 8; ++r)
            sE[(m0 + r + lh * 8) * 64 + n0 + lm] = acc[r];
    }
    __syncthreads();

    // ---- per-channel argmax over k of ||h2(:,c)||, transposed store ----
    if (t < 64) {
        int c = t;
        float best = -1.0f;
        int bk = 0;
        for (int k = 0; k < K_; ++k) {
            float a0 = sE[(3 * k) * 64 + c];
            float a1 = sE[(3 * k + 1) * 64 + c];
            float a2 = sE[(3 * k + 2) * 64 + c];
            float nn = a0 * a0 + a1 * a1 + a2 * a2;
            if (nn > best) { best = nn; bk = k; }
        }
        size_t ob = ((size_t)bn * 64 + c) * 3;
        out[ob + 0] = sE[(3 * bk) * 64 + c];
        out[ob + 1] = sE[(3 * bk + 1) * 64 + c];
        out[ob + 2] = sE[(3 * bk + 2) * 64 + c];
    }
}

extern "C" void kernel_launch(void* const* d_in, const int* in_sizes, int n_in,
                              void* d_out, int out_size, void* d_ws, size_t ws_size,
                              hipStream_t stream) {
    const float* x  = (const float*)d_in[0];   // (4,2048,3,64)
    const float* W1 = (const float*)d_in[1];   // (128,64)
    const float* W2 = (const float*)d_in[2];   // (64,64)
    float* out = (float*)d_out;                // (4,2048,64,3)

    float* pos = (float*)d_ws;                                   // 24576 floats
    int* knn = (int*)((char*)d_ws + (size_t)B_ * N_ * 3 * 4);    // 8192*20 ints

    pos_kernel<<<(B_ * N_ * 3 + 255) / 256, 256, 0, stream>>>(x, pos);
    knn_kernel<<<64, 128, 0, stream>>>(pos, knn);
    fused_kernel<<<B_ * N_, 128, 0, stream>>>(x, W1, W2, knn, out);
}